// GNNWithAttention_88098369175981
// MI455X (gfx1250) — compile-verified
//
#include <hip/hip_runtime.h>
#include <hip/hip_bf16.h>
#include <math.h>

typedef float v2f __attribute__((ext_vector_type(2)));
typedef float v8f __attribute__((ext_vector_type(8)));

#define NEG_SLOPE 0.2f

// edge e in [0, E): real edge; e in [E, E+N): self loop (e-E, e-E)
__device__ __forceinline__ void edge_sd(int e, int E, const int* __restrict__ ei,
                                        int& s, int& d) {
    if (e < E) { s = ei[e]; d = ei[E + e]; }
    else       { s = e - E; d = s; }
}

__device__ __forceinline__ void atomicMaxF(float* addr, float val) {
    int old = __float_as_int(*addr);
    while (val > __int_as_float(old)) {
        int assumed = old;
        old = atomicCAS((int*)addr, assumed, __float_as_int(val));
        if (old == assumed) break;
    }
}

__global__ void fill_kernel(float* __restrict__ p, float v, int n) {
    int i = blockIdx.x * blockDim.x + threadIdx.x;
    if (i < n) p[i] = v;
}

// C[M x Ncols] = A[M x K] * B[K x Ncols], all row-major fp32.
// One wave per 16x16 output tile, V_WMMA_F32_16X16X4_F32 over K.
// Requires M%16==0, Ncols%16==0, K%4==0.
__global__ void wmma_gemm_f32(const float* __restrict__ A, const float* __restrict__ B,
                              float* __restrict__ C, int M, int K, int Ncols) {
    int wid  = (blockIdx.x * blockDim.x + threadIdx.x) >> 5;
    int lane = threadIdx.x & 31;
    int tiles_n = Ncols >> 4;
    int tiles_total = (M >> 4) * tiles_n;
    if (wid >= tiles_total) return;          // whole wave exits: EXEC stays all-1s
    int row0 = (wid / tiles_n) << 4;
    int col0 = (wid % tiles_n) << 4;

    int mi = lane & 15;                      // M (for A) / N (for B,C) index
    int kh = (lane >> 4) << 1;               // lanes 16-31 hold K+2,K+3

    const float* arow = A + (size_t)(row0 + mi) * K;
    const float* bcol = B + col0 + mi;

    v8f acc = {};
    for (int k = 0; k < K; k += 4) {
        v2f a, b;
        a.x = arow[k + kh];                          // A[m][k+kh+0]
        a.y = arow[k + kh + 1];                      // A[m][k+kh+1]
        b.x = bcol[(size_t)(k + kh) * Ncols];        // B[k+kh+0][n]
        b.y = bcol[(size_t)(k + kh + 1) * Ncols];    // B[k+kh+1][n]
        acc = __builtin_amdgcn_wmma_f32_16x16x4_f32(
            false, a, false, b, (short)0, acc, false, false);
    }
    // D: VGPR j holds row M = 8*(lane>>4) + j, column N = mi
    int mbase = row0 + ((lane >> 4) << 3);
    float* crow = C + (size_t)mbase * Ncols + col0 + mi;
#pragma unroll
    for (int j = 0; j < 8; ++j) crow[(size_t)j * Ncols] = acc[j];
}

// a_s[n,h] = sum_c h[n, h*C+c]*att_s[h,c];  a_d likewise. One thread per (n,h).
__global__ void scores_kernel(const float* __restrict__ h,
                              const float* __restrict__ att_s,
                              const float* __restrict__ att_d,
                              float* __restrict__ a_s, float* __restrict__ a_d,
                              int N, int H, int C) {
    int t = blockIdx.x * blockDim.x + threadIdx.x;
    if (t >= N * H) return;
    int hh = t % H;
    const float* hp = h + (size_t)(t / H) * H * C + (size_t)hh * C;
    const float* sp = att_s + (size_t)hh * C;
    const float* dp = att_d + (size_t)hh * C;
    float ss = 0.f, dd = 0.f;
    for (int c = 0; c < C; ++c) { float v = hp[c]; ss += v * sp[c]; dd += v * dp[c]; }
    a_s[t] = ss; a_d[t] = dd;
}

// Pass 1: segment max of leaky_relu scores into emax[dst,h]
__global__ void edge_max_kernel(const int* __restrict__ ei, int E, int ET,
                                const float* __restrict__ a_s, const float* __restrict__ a_d,
                                float* __restrict__ emax, int H) {
    int t = blockIdx.x * blockDim.x + threadIdx.x;
    if (t >= ET * H) return;
    int e = t / H, hh = t % H;
    int s, d; edge_sd(e, E, ei, s, d);
    float x = a_s[s * H + hh] + a_d[d * H + hh];
    x = (x > 0.f) ? x : NEG_SLOPE * x;
    atomicMaxF(&emax[d * H + hh], x);
}

// Pass 2: p = exp(e - emax[dst]); store p per edge; denom[dst] += p
__global__ void edge_sum_kernel(const int* __restrict__ ei, int E, int ET,
                                const float* __restrict__ a_s, const float* __restrict__ a_d,
                                const float* __restrict__ emax,
                                float* __restrict__ pbuf, float* __restrict__ denom, int H) {
    int t = blockIdx.x * blockDim.x + threadIdx.x;
    if (t >= ET * H) return;
    int e = t / H, hh = t % H;
    int s, d; edge_sd(e, E, ei, s, d);
    float x = a_s[s * H + hh] + a_d[d * H + hh];
    x = (x > 0.f) ? x : NEG_SLOPE * x;
    float p = expf(x - emax[d * H + hh]);
    pbuf[t] = p;
    atomicAdd(&denom[d * H + hh], p);
}

// Pass 3: agg[dst, h, :] += (p/denom) * hfeat[src, h, :]. One wave per edge.
__global__ void edge_agg_kernel(const int* __restrict__ ei, int E, int ET,
                                const float* __restrict__ pbuf, const float* __restrict__ denom,
                                const float* __restrict__ hfeat, float* __restrict__ agg,
                                int H, int C) {
    int wid  = (blockIdx.x * blockDim.x + threadIdx.x) >> 5;
    int lane = threadIdx.x & 31;
    if (wid >= ET) return;
    int s, d; edge_sd(wid, E, ei, s, d);
    int HC = H * C;
    const float* hs = hfeat + (size_t)s * HC;
    float*       ag = agg   + (size_t)d * HC;
    for (int c = lane; c < HC; c += 32) {
        int hh = c / C;  // C is a power of two -> shift
        float alpha = pbuf[wid * H + hh] / (denom[d * H + hh] + 1e-16f);
        atomicAdd(&ag[c], hs[c] * alpha);
    }
}

// h = elu(agg + bias) elementwise (in place)
__global__ void elu_bias_kernel(float* __restrict__ a, const float* __restrict__ bias,
                                int total, int cols) {
    int i = blockIdx.x * blockDim.x + threadIdx.x;
    if (i >= total) return;
    float v = a[i] + bias[i % cols];
    a[i] = (v > 0.f) ? v : expm1f(v);
}

// out[n, 0:2] = (agg2[n,:] + b2) @ fcW + fcb
__global__ void final_fc_kernel(const float* __restrict__ agg2, const float* __restrict__ b2,
                                const float* __restrict__ fcW, const float* __restrict__ fcb,
                                float* __restrict__ out, int N, int C) {
    int n = blockIdx.x * blockDim.x + threadIdx.x;
    if (n >= N) return;
    float y0 = fcb[0], y1 = fcb[1];
    const float* a = agg2 + (size_t)n * C;
    for (int c = 0; c < C; ++c) {
        float v = a[c] + b2[c];
        y0 += v * fcW[c * 2 + 0];
        y1 += v * fcW[c * 2 + 1];
    }
    out[n * 2 + 0] = y0;
    out[n * 2 + 1] = y1;
}

extern "C" void kernel_launch(void* const* d_in, const int* in_sizes, int n_in,
                              void* d_out, int out_size, void* d_ws, size_t ws_size,
                              hipStream_t stream) {
    const float* x      = (const float*)d_in[0];
    const int*   ei     = (const int*)  d_in[1];
    const float* W1     = (const float*)d_in[2];
    const float* att_s1 = (const float*)d_in[3];
    const float* att_d1 = (const float*)d_in[4];
    const float* b1     = (const float*)d_in[5];
    const float* W2     = (const float*)d_in[6];
    const float* att_s2 = (const float*)d_in[7];
    const float* att_d2 = (const float*)d_in[8];
    const float* b2     = (const float*)d_in[9];
    const float* fcW    = (const float*)d_in[10];
    const float* fcb    = (const float*)d_in[11];
    float* out = (float*)d_out;
    (void)n_in; (void)out_size; (void)ws_size;

    const int F_IN = 128, HID = 64, HEADS = 4, F_OUT = 64;
    const int HC1 = HEADS * HID;            // 256
    const int N  = in_sizes[0] / F_IN;      // 50000
    const int E  = in_sizes[1] / 2;         // 800000
    const int ET = E + N;                   // + self loops

    // workspace carve (256B aligned)
    char* wsb = (char*)d_ws;
    size_t off = 0;
    auto carve = [&](size_t elems) -> float* {
        float* p = (float*)(wsb + off);
        off += ((elems * sizeof(float)) + 255) & ~(size_t)255;
        return p;
    };
    float* h1    = carve((size_t)N * HC1);      // x@W1
    float* agg1  = carve((size_t)N * HC1);      // aggregated -> elu in place
    float* as1   = carve((size_t)N * HEADS);
    float* ad1   = carve((size_t)N * HEADS);
    float* emax1 = carve((size_t)N * HEADS);
    float* den1  = carve((size_t)N * HEADS);
    float* p1    = carve((size_t)ET * HEADS);
    float* h2    = carve((size_t)N * F_OUT);    // agg1@W2
    float* agg2  = carve((size_t)N * F_OUT);
    float* as2   = carve((size_t)N);
    float* ad2   = carve((size_t)N);
    float* emax2 = carve((size_t)N);
    float* den2  = carve((size_t)N);
    float* p2    = carve((size_t)ET);

    const int BT = 256;
    auto nb = [](long long t, int b) { return (int)((t + b - 1) / b); };

    // ---------------- layer 1 (H=4, C=64, concat) ----------------
    fill_kernel<<<nb((long long)N * HC1, BT), BT, 0, stream>>>(agg1, 0.0f, N * HC1);
    fill_kernel<<<nb((long long)N * HEADS, BT), BT, 0, stream>>>(emax1, -INFINITY, N * HEADS);
    fill_kernel<<<nb((long long)N * HEADS, BT), BT, 0, stream>>>(den1, 0.0f, N * HEADS);

    {   // h1 = x @ W1   (50000x128 * 128x256)
        long long waves = (long long)(N / 16) * (HC1 / 16);
        wmma_gemm_f32<<<nb(waves * 32, BT), BT, 0, stream>>>(x, W1, h1, N, F_IN, HC1);
    }
    scores_kernel<<<nb((long long)N * HEADS, BT), BT, 0, stream>>>(
        h1, att_s1, att_d1, as1, ad1, N, HEADS, HID);
    edge_max_kernel<<<nb((long long)ET * HEADS, BT), BT, 0, stream>>>(
        ei, E, ET, as1, ad1, emax1, HEADS);
    edge_sum_kernel<<<nb((long long)ET * HEADS, BT), BT, 0, stream>>>(
        ei, E, ET, as1, ad1, emax1, p1, den1, HEADS);
    edge_agg_kernel<<<nb((long long)ET * 32, BT), BT, 0, stream>>>(
        ei, E, ET, p1, den1, h1, agg1, HEADS, HID);
    elu_bias_kernel<<<nb((long long)N * HC1, BT), BT, 0, stream>>>(agg1, b1, N * HC1, HC1);

    // ---------------- layer 2 (H=1, C=64, mean==identity) ----------------
    fill_kernel<<<nb((long long)N * F_OUT, BT), BT, 0, stream>>>(agg2, 0.0f, N * F_OUT);
    fill_kernel<<<nb((long long)N, BT), BT, 0, stream>>>(emax2, -INFINITY, N);
    fill_kernel<<<nb((long long)N, BT), BT, 0, stream>>>(den2, 0.0f, N);

    {   // h2 = elu_out @ W2  (50000x256 * 256x64)
        long long waves = (long long)(N / 16) * (F_OUT / 16);
        wmma_gemm_f32<<<nb(waves * 32, BT), BT, 0, stream>>>(agg1, W2, h2, N, HC1, F_OUT);
    }
    scores_kernel<<<nb((long long)N, BT), BT, 0, stream>>>(
        h2, att_s2, att_d2, as2, ad2, N, 1, F_OUT);
    edge_max_kernel<<<nb((long long)ET, BT), BT, 0, stream>>>(
        ei, E, ET, as2, ad2, emax2, 1);
    edge_sum_kernel<<<nb((long long)ET, BT), BT, 0, stream>>>(
        ei, E, ET, as2, ad2, emax2, p2, den2, 1);
    edge_agg_kernel<<<nb((long long)ET * 32, BT), BT, 0, stream>>>(
        ei, E, ET, p2, den2, h2, agg2, 1, F_OUT);

    // ---------------- final FC ----------------
    final_fc_kernel<<<nb((long long)N, BT), BT, 0, stream>>>(
        agg2, b2, fcW, fcb, out, N, F_OUT);
}